// HistogramBlock_31799937859956
// MI455X (gfx1250) — compile-verified
//
#include <hip/hip_runtime.h>
#include <hip/hip_bf16.h>

#define NBINS   256
#define NPLANE  96          // 32 * 3
#define HOUT    512
#define WOUT    512
#define PLANE_ELEMS (512 * 512)

typedef __attribute__((ext_vector_type(2))) float v2f;
typedef __attribute__((ext_vector_type(8))) float v8f;

// ---------------------------------------------------------------------------
// Kernel 1: zero the global histograms (d_ws is poisoned between runs)
// ---------------------------------------------------------------------------
__global__ void zero_hist_kernel(unsigned* __restrict__ gh) {
    gh[blockIdx.x * blockDim.x + threadIdx.x] = 0u;
}

// ---------------------------------------------------------------------------
// Kernel 2: histogram. Read-bandwidth bound (100.7 MB).
// grid = (32 chunks, 96 planes), block = 256. Each block: 8192 floats via
// float4 loads -> LDS histogram (ds_add_u32) -> one global atomic per bin.
// ---------------------------------------------------------------------------
__global__ void hist_kernel(const float* __restrict__ x, unsigned* __restrict__ gh) {
    __shared__ unsigned lh[NBINS];
    const int plane = blockIdx.y;
    const int chunk = blockIdx.x;

    lh[threadIdx.x] = 0u;
    __syncthreads();

    const float4* __restrict__ p4 =
        (const float4*)(x + (size_t)plane * PLANE_ELEMS + (size_t)chunk * 8192);

#pragma unroll
    for (int i = 0; i < 8; ++i) {
        float4 v = p4[i * 256 + threadIdx.x];
        // torch.histc semantics: valid iff 0<=v<=1, bin=floor(v*256), v==1 -> 255
        float vals[4] = {v.x, v.y, v.z, v.w};
#pragma unroll
        for (int j = 0; j < 4; ++j) {
            float f = vals[j];
            if (f >= 0.0f && f <= 1.0f) {
                int b = (int)(f * 256.0f);     // f >= 0 -> trunc == floor
                if (b > 255) b = 255;
                atomicAdd(&lh[b], 1u);
            }
        }
    }
    __syncthreads();
    atomicAdd(&gh[plane * NBINS + threadIdx.x], lh[threadIdx.x]);
}

// ---------------------------------------------------------------------------
// Kernel 3: bilinear upsample expressed as Up[512,96] = M[512,256] x H[256,96]
// on V_WMMA_F32_16X16X4_F32. One wave per 16x16 output tile, 64 K-steps.
// f32 WMMA is the *required* precision: counts reach 2^18 (fp16 overflows),
// and with weights in {0.25,0.75,1.0} the result is bit-exact vs reference.
// grid = (32 h-tiles, 6 plane-tiles), block = 32 (one wave, EXEC all-ones).
// ---------------------------------------------------------------------------
__global__ void interp_wmma_kernel(const unsigned* __restrict__ gh,
                                   float* __restrict__ up) {
    const int lane   = threadIdx.x;      // 0..31
    const int half   = lane >> 4;        // 0: lanes 0-15, 1: lanes 16-31
    const int idx    = lane & 15;
    const int tile_m = blockIdx.x;       // h tile   (0..31)
    const int tile_n = blockIdx.y;       // plane tile (0..5)

    // Interpolation weights for output row h (A-matrix row, M = idx per ISA layout)
    const int   h  = tile_m * 16 + idx;
    float sy = 0.5f * (float)h - 0.25f;
    if (sy < 0.0f) sy = 0.0f;
    const int   y0 = (int)sy;
    int         y1 = y0 + 1; if (y1 > NBINS - 1) y1 = NBINS - 1;
    const float t  = sy - (float)y0;
    const float w0 = 1.0f - t;

    // B-matrix column (N = idx per ISA layout)
    const int p = tile_n * 16 + idx;
    const unsigned* __restrict__ hp = gh + (size_t)p * NBINS;

    v8f c = {0.f, 0.f, 0.f, 0.f, 0.f, 0.f, 0.f, 0.f};

    for (int k0 = 0; k0 < NBINS; k0 += 4) {
        // A 16x4 layout: VGPR0 = K (k0 + 2*half), VGPR1 = K+1  (ISA §7.12.2)
        const int ka = k0 + 2 * half;
        const int kb = ka + 1;
        v2f a, b;
        a.x = ((ka == y0) ? w0 : 0.0f) + ((ka == y1) ? t : 0.0f);
        a.y = ((kb == y0) ? w0 : 0.0f) + ((kb == y1) ? t : 0.0f);
        // B 4x16 layout mirrors A: row K striped across lanes within a VGPR
        b.x = (float)hp[ka];
        b.y = (float)hp[kb];
        c = __builtin_amdgcn_wmma_f32_16x16x4_f32(
                /*neg_a=*/false, a, /*neg_b=*/false, b,
                /*c_mod=*/(short)0, c, /*reuse_a=*/false, /*reuse_b=*/false);
    }

    // C/D 16x16 layout: VGPR r -> M = r + 8*half, N = idx
#pragma unroll
    for (int r = 0; r < 8; ++r) {
        const int hh = tile_m * 16 + r + 8 * half;
        up[(size_t)(tile_n * 16 + idx) * HOUT + hh] = c[r];
    }
}

// ---------------------------------------------------------------------------
// Kernel 4: broadcast along W. Store-bandwidth bound (100.7 MB).
// grid = 96*512 rows, block = 128; each lane stores one float4 (b128 store).
// ---------------------------------------------------------------------------
__global__ void bcast_kernel(const float* __restrict__ up, float* __restrict__ out) {
    const int row = blockIdx.x;                 // plane*512 + h
    const float v = up[row];
    const float4 vv = {v, v, v, v};
    ((float4*)(out + (size_t)row * WOUT))[threadIdx.x] = vv;
}

// ---------------------------------------------------------------------------
extern "C" void kernel_launch(void* const* d_in, const int* in_sizes, int n_in,
                              void* d_out, int out_size, void* d_ws, size_t ws_size,
                              hipStream_t stream) {
    const float* x   = (const float*)d_in[0];
    float*       out = (float*)d_out;

    unsigned* gh = (unsigned*)d_ws;                                   // 96*256 u32
    float*    up = (float*)((char*)d_ws + (size_t)NPLANE * NBINS * 4); // 96*512 f32

    zero_hist_kernel<<<NPLANE, NBINS, 0, stream>>>(gh);
    hist_kernel<<<dim3(32, NPLANE), 256, 0, stream>>>(x, gh);
    interp_wmma_kernel<<<dim3(HOUT / 16, NPLANE / 16), 32, 0, stream>>>(gh, up);
    bcast_kernel<<<NPLANE * HOUT, WOUT / 4, 0, stream>>>(up, out);
}